// LatentCube_40853728919606
// MI455X (gfx1250) — compile-verified
//
#include <hip/hip_runtime.h>
#include <hip/hip_bf16.h>

// Problem constants (from reference)
#define B_      16
#define LAT_    4096
#define NF_     64
#define CUBE_   16
#define VOX_    4096      // 16^3
#define CF_     7         // 1 + 3*len(KS)
#define FIN_    448       // NF * CF
#define FH_     64
#define FOUT_   64
#define TOTAL_  65728
#define N_CALLS_ 4
#define NT_     64        // voxels per MLP block
#define PAD_    16        // LDS row pad (keeps 16B alignment, splits bank groups)

// ks per-batch row offsets (cumsum of SIZES)
#define OFF_WIN   0
#define OFF_BIN   28672
#define OFF_WMID  28736
#define OFF_BMID  32832
#define OFF_WOUT  32896
#define OFF_BOUT  36992
#define OFF_WSK   37056

#if defined(__has_builtin)
#  if __has_builtin(__builtin_amdgcn_global_load_async_to_lds_b128)
#    define HAVE_ASYNC_LDS 1
#  endif
#endif
#ifndef HAVE_ASYNC_LDS
#  define HAVE_ASYNC_LDS 0
#endif

typedef float v2f __attribute__((ext_vector_type(2)));
typedef float v8f __attribute__((ext_vector_type(8)));

#if HAVE_ASYNC_LDS
// Builtin signature (from compiler diagnostic): param is V4i (vector_size 16)
// pointers: (global int4*, lds int4*, imm offset, imm cpol).
typedef int async_i4 __attribute__((vector_size(16)));
typedef __attribute__((address_space(1))) async_i4* async_gp;
typedef __attribute__((address_space(3))) async_i4* async_lp;
#endif

// V_WMMA_F32_16X16X4_F32: D(16x16 f32) = A(16x4 f32) * B(4x16 f32) + C
// A: vgpr0 = K{0|2}, vgpr1 = K{1|3} (lanes 0-15: K0/K1, lanes 16-31: K2/K3), M = lane&15
// B: mirrored with N = lane&15
// C/D: vgpr r -> M = r + 8*(lane>=16), N = lane&15
static __device__ inline v8f wmma4(v2f a, v2f b, v8f c) {
  return __builtin_amdgcn_wmma_f32_16x16x4_f32(
      /*neg_a=*/false, a, /*neg_b=*/false, b,
      /*c_mod=*/(short)0, c, /*reuse_a=*/false, /*reuse_b=*/false);
}

// ---------------------------------------------------------------------------
// Kernel 1: ks[16][TOTAL] = lat(16x4096) @ w_hyper(4096xTOTAL) + b_hyper
// One wave per 32-wide column tile (two WMMA N-tiles share one A-fragment):
// each half-wave touches exactly one aligned 128B line per w_hyper row, so
// the dominant 1.077 GB stream is fetched once at full line utilization.
// Computed ONCE (loop-invariant) instead of 4x like the reference.
// ---------------------------------------------------------------------------
__global__ __launch_bounds__(32)
void hyper_gemm_kernel(const float* __restrict__ lat,
                       const float* __restrict__ w_hyper,
                       const float* __restrict__ b_hyper,
                       float* __restrict__ ks) {
  const int n0   = blockIdx.x * 32;
  const int lane = threadIdx.x;
  const int half = lane >> 4;
  const int m    = lane & 15;

  v8f acc0 = {}, acc1 = {};
  const float* latrow = lat + (size_t)m * LAT_;
  #pragma unroll 16
  for (int kk = 0; kk < LAT_; kk += 4) {
    v2f a, b0, b1;
    a.x = latrow[kk + 2 * half];
    a.y = latrow[kk + 2 * half + 1];
    const size_t r0 = (size_t)(kk + 2 * half) * TOTAL_ + (size_t)(n0 + m);
    b0.x = w_hyper[r0];
    b0.y = w_hyper[r0 + TOTAL_];
    b1.x = w_hyper[r0 + 16];
    b1.y = w_hyper[r0 + 16 + TOTAL_];
    if ((kk & 63) == 0 && kk + 64 < LAT_) {
      // keep the HBM stream deep: prefetch 64 rows ahead (global_prefetch_b8)
      __builtin_prefetch(&w_hyper[r0 + (size_t)64 * TOTAL_], 0, 0);
    }
    acc0 = wmma4(a, b0, acc0);
    acc1 = wmma4(a, b1, acc1);
  }
  const float bias0 = b_hyper[n0 + m];
  const float bias1 = b_hyper[n0 + 16 + m];
  #pragma unroll
  for (int r = 0; r < 8; ++r) {
    const int row = r + 8 * half;        // batch index 0..15
    ks[(size_t)row * TOTAL_ + (size_t)(n0 + m)]      = acc0[r] + bias0;
    ks[(size_t)row * TOTAL_ + (size_t)(n0 + 16 + m)] = acc1[r] + bias1;
  }
}

// ---------------------------------------------------------------------------
// Kernel 2: state[b][c][v] = seed[c][v]  (tile seed across batch)
// ---------------------------------------------------------------------------
__global__ __launch_bounds__(256)
void init_state_kernel(const float* __restrict__ seed, float* __restrict__ state) {
  int i = blockIdx.x * 256 + threadIdx.x;
  if (i < B_ * NF_ * VOX_) state[i] = seed[i % (NF_ * VOX_)];
}

// ---------------------------------------------------------------------------
// Kernel 3: x = instance_norm(sin([state, sobel3(state), sobel5(state)]))
// One block per (b, c). Channel volume (16KB) cached in LDS; 7 feature
// channels with block-wide mean/var reductions.
// Cross-correlation (lax conv), zero padding, weight = u/(dx^2+dy^2+dz^2).
// Feature index: f==0 -> c ; else 64 + 192*jk + 3*c + dir (matches concat +
// np.tile channel ordering with feature_group_count=C).
// ---------------------------------------------------------------------------
__global__ __launch_bounds__(256)
void sobel_sin_norm_kernel(const float* __restrict__ state, float* __restrict__ x) {
  __shared__ float tile[VOX_];
  __shared__ float redA[256];
  __shared__ float redB[256];

  const int b   = blockIdx.x / NF_;
  const int c   = blockIdx.x % NF_;
  const int tid = threadIdx.x;

  const float* src = state + ((size_t)b * NF_ + c) * VOX_;
  for (int i = tid; i < VOX_; i += 256) tile[i] = src[i];
  __syncthreads();

  for (int f = 0; f < CF_; ++f) {
    const int jk  = (f - 1) / 3;          // 0 -> k=3, 1 -> k=5 (unused for f==0)
    const int dir = (f - 1) % 3;
    const int r   = jk ? 2 : 1;

    float vreg[16];
    float s = 0.f, s2 = 0.f;

    for (int j = 0; j < 16; ++j) {
      const int v  = tid + 256 * j;
      float val;
      if (f == 0) {
        val = sinf(tile[v]);
      } else {
        const int zz = v >> 8, yy = (v >> 4) & 15, xx = v & 15;
        float acc = 0.f;
        for (int a = -r; a <= r; ++a) {
          const int z2 = zz + a; if (z2 < 0 || z2 > 15) continue;
          for (int bb = -r; bb <= r; ++bb) {
            const int y2 = yy + bb; if (y2 < 0 || y2 > 15) continue;
            for (int cc = -r; cc <= r; ++cc) {
              const int x2 = xx + cc; if (x2 < 0 || x2 > 15) continue;
              const int u = (dir == 0) ? a : (dir == 1) ? bb : cc;
              if (u == 0) continue;          // zero weight (covers center too)
              const int den = a * a + bb * bb + cc * cc;
              acc += ((float)u / (float)den) * tile[(z2 << 8) | (y2 << 4) | x2];
            }
          }
        }
        val = sinf(acc);
      }
      vreg[j] = val; s += val; s2 += val * val;
    }

    // block reduction for mean / var
    redA[tid] = s; redB[tid] = s2;
    __syncthreads();
    for (int off = 128; off > 0; off >>= 1) {
      if (tid < off) { redA[tid] += redA[tid + off]; redB[tid] += redB[tid + off]; }
      __syncthreads();
    }
    const float mean  = redA[0] * (1.f / VOX_);
    const float var   = redB[0] * (1.f / VOX_) - mean * mean;
    const float scale = rsqrtf(var + 1e-5f);
    __syncthreads();   // protect redA/redB before next feature reuses them

    const int ff = (f == 0) ? c : (NF_ + 192 * jk + 3 * c + dir);
    float* dst = x + ((size_t)b * FIN_ + ff) * VOX_;
    for (int j = 0; j < 16; ++j) dst[tid + 256 * j] = (vreg[j] - mean) * scale;
  }
}

// ---------------------------------------------------------------------------
// Kernel 4: fused dynamic MLP + residual update.
// Block = 4 waves (128 thr), one 64-voxel N-tile of one batch. Wave w owns
// output-channel rows [16w, 16w+16) x all 64 columns (4 WMMA N-tiles sharing
// each A-fragment -> 4x weight-load reuse). x tile (448x64) staged into LDS
// with CDNA5 async-to-LDS b128 when available; h1/h2 never leave LDS.
// All four GEMMs are V_WMMA_F32_16X16X4_F32 chains. state += 0.1 * res.
// LDS: 448*80 + 2*64*80 floats = 180 KB (< 320 KB per WGP).
// ---------------------------------------------------------------------------
__global__ __launch_bounds__(128)
void mlp_kernel(const float* __restrict__ ks, const float* __restrict__ x,
                float* __restrict__ state) {
  __shared__ float x_s[FIN_][NT_ + PAD_];
  __shared__ float h1_s[FH_][NT_ + PAD_];
  __shared__ float h2_s[FH_][NT_ + PAD_];

  const int b    = blockIdx.x / (VOX_ / NT_);
  const int v0   = (blockIdx.x % (VOX_ / NT_)) * NT_;
  const int tid  = threadIdx.x;
  const int wave = tid >> 5;
  const int lane = tid & 31;
  const int half = lane >> 4;
  const int ln   = lane & 15;
  const int m0   = wave * 16;

  const float* kb = ks + (size_t)b * TOTAL_;
  const float* xb = x + (size_t)b * FIN_ * VOX_;

  // ---- stage x tile (448 x 64) into LDS, 16B chunks ----
#if HAVE_ASYNC_LDS
  for (int i = tid; i < FIN_ * (NT_ / 4); i += 128) {
    const int k = i >> 4, j = i & 15;      // NT_/4 == 16 chunks per row
    __builtin_amdgcn_global_load_async_to_lds_b128(
        (async_gp)(xb + (size_t)k * VOX_ + v0 + 4 * j),
        (async_lp)&x_s[k][4 * j], 0, 0);
  }
  asm volatile("s_wait_asynccnt 0" ::: "memory");
#else
  for (int i = tid; i < FIN_ * (NT_ / 4); i += 128) {
    const int k = i >> 4, j = i & 15;
    *(float4*)&x_s[k][4 * j] = *(const float4*)(xb + (size_t)k * VOX_ + v0 + 4 * j);
  }
#endif
  __syncthreads();

  // ---- h1 = relu(w_in @ x + b_in) ----
  {
    v8f acc[4] = {};
    const float* w = kb + OFF_WIN + (size_t)(m0 + ln) * FIN_;
    #pragma unroll 4
    for (int kk = 0; kk < FIN_; kk += 4) {
      v2f a;
      a.x = w[kk + 2 * half];  a.y = w[kk + 2 * half + 1];
      const float* xr0 = &x_s[kk + 2 * half][0];
      const float* xr1 = &x_s[kk + 2 * half + 1][0];
      #pragma unroll
      for (int ns = 0; ns < 4; ++ns) {
        v2f bf; bf.x = xr0[ns * 16 + ln]; bf.y = xr1[ns * 16 + ln];
        acc[ns] = wmma4(a, bf, acc[ns]);
      }
    }
    #pragma unroll
    for (int r = 0; r < 8; ++r) {
      const int row = m0 + r + 8 * half;
      const float bias = kb[OFF_BIN + row];
      #pragma unroll
      for (int ns = 0; ns < 4; ++ns)
        h1_s[row][ns * 16 + ln] = fmaxf(acc[ns][r] + bias, 0.f);
    }
  }
  __syncthreads();

  // ---- h2 = relu(w_mid @ h1 + b_mid) ----
  {
    v8f acc[4] = {};
    const float* w = kb + OFF_WMID + (size_t)(m0 + ln) * FH_;
    #pragma unroll 4
    for (int kk = 0; kk < FH_; kk += 4) {
      v2f a;
      a.x = w[kk + 2 * half];  a.y = w[kk + 2 * half + 1];
      const float* hr0 = &h1_s[kk + 2 * half][0];
      const float* hr1 = &h1_s[kk + 2 * half + 1][0];
      #pragma unroll
      for (int ns = 0; ns < 4; ++ns) {
        v2f bf; bf.x = hr0[ns * 16 + ln]; bf.y = hr1[ns * 16 + ln];
        acc[ns] = wmma4(a, bf, acc[ns]);
      }
    }
    #pragma unroll
    for (int r = 0; r < 8; ++r) {
      const int row = m0 + r + 8 * half;
      const float bias = kb[OFF_BMID + row];
      #pragma unroll
      for (int ns = 0; ns < 4; ++ns)
        h2_s[row][ns * 16 + ln] = fmaxf(acc[ns][r] + bias, 0.f);
    }
  }
  __syncthreads();

  // ---- h3 = w_out @ h2 + b_out ; skip = w_sk @ x ; state += 0.1*(h3+skip) ----
  {
    v8f acc[4] = {};
    const float* w = kb + OFF_WOUT + (size_t)(m0 + ln) * FH_;
    #pragma unroll 4
    for (int kk = 0; kk < FH_; kk += 4) {
      v2f a;
      a.x = w[kk + 2 * half];  a.y = w[kk + 2 * half + 1];
      const float* hr0 = &h2_s[kk + 2 * half][0];
      const float* hr1 = &h2_s[kk + 2 * half + 1][0];
      #pragma unroll
      for (int ns = 0; ns < 4; ++ns) {
        v2f bf; bf.x = hr0[ns * 16 + ln]; bf.y = hr1[ns * 16 + ln];
        acc[ns] = wmma4(a, bf, acc[ns]);
      }
    }
    v8f sk[4] = {};
    const float* ws = kb + OFF_WSK + (size_t)(m0 + ln) * FIN_;
    #pragma unroll 4
    for (int kk = 0; kk < FIN_; kk += 4) {
      v2f a;
      a.x = ws[kk + 2 * half];  a.y = ws[kk + 2 * half + 1];
      const float* xr0 = &x_s[kk + 2 * half][0];
      const float* xr1 = &x_s[kk + 2 * half + 1][0];
      #pragma unroll
      for (int ns = 0; ns < 4; ++ns) {
        v2f bf; bf.x = xr0[ns * 16 + ln]; bf.y = xr1[ns * 16 + ln];
        sk[ns] = wmma4(a, bf, sk[ns]);
      }
    }
    #pragma unroll
    for (int r = 0; r < 8; ++r) {
      const int row = m0 + r + 8 * half;
      const float bias = kb[OFF_BOUT + row];
      #pragma unroll
      for (int ns = 0; ns < 4; ++ns) {
        const float res = acc[ns][r] + bias + sk[ns][r];
        const size_t idx = ((size_t)b * NF_ + row) * VOX_ + v0 + ns * 16 + ln;
        state[idx] += 0.1f * res;
      }
    }
  }
}

// ---------------------------------------------------------------------------
// Kernel 5: out[b][v] = mean_c state[b][c][v]   (B*VOX == 16*4096 outputs)
// ---------------------------------------------------------------------------
__global__ __launch_bounds__(256)
void mean_kernel(const float* __restrict__ state, float* __restrict__ out) {
  const int i = blockIdx.x * 256 + threadIdx.x;
  if (i >= B_ * VOX_) return;
  const int b = i / VOX_, v = i % VOX_;
  const float* s = state + (size_t)b * NF_ * VOX_ + v;
  float acc = 0.f;
  #pragma unroll 8
  for (int c = 0; c < NF_; ++c) acc += s[(size_t)c * VOX_];
  out[i] = acc * (1.f / NF_);
}

// ---------------------------------------------------------------------------
extern "C" void kernel_launch(void* const* d_in, const int* in_sizes, int n_in,
                              void* d_out, int out_size, void* d_ws, size_t ws_size,
                              hipStream_t stream) {
  const float* lat     = (const float*)d_in[0];
  const float* seed    = (const float*)d_in[1];
  const float* w_hyper = (const float*)d_in[2];
  const float* b_hyper = (const float*)d_in[3];

  // Workspace layout (fp32): ks | state | x   (~138 MB total)
  float* ks    = (float*)d_ws;
  float* state = ks + (size_t)B_ * TOTAL_;
  float* x     = state + (size_t)B_ * NF_ * VOX_;

  // Loop-invariant hypernetwork GEMM: done once (reference redoes it 4x).
  hyper_gemm_kernel<<<TOTAL_ / 32, 32, 0, stream>>>(lat, w_hyper, b_hyper, ks);
  init_state_kernel<<<(B_ * NF_ * VOX_ + 255) / 256, 256, 0, stream>>>(seed, state);

  for (int it = 0; it < N_CALLS_; ++it) {
    sobel_sin_norm_kernel<<<B_ * NF_, 256, 0, stream>>>(state, x);
    mlp_kernel<<<B_ * (VOX_ / NT_), 128, 0, stream>>>(ks, x, state);
  }

  mean_kernel<<<(B_ * VOX_ + 255) / 256, 256, 0, stream>>>(state, (float*)d_out);
}